// TemporalGNN_83322365542776
// MI455X (gfx1250) — compile-verified
//
#include <hip/hip_runtime.h>
#include <math.h>

// ---------------------------------------------------------------------------
// Types / WMMA
// ---------------------------------------------------------------------------
typedef __attribute__((ext_vector_type(2))) float v2f;
typedef __attribute__((ext_vector_type(8))) float v8f;

#if defined(__gfx1250__)
#if __has_builtin(__builtin_amdgcn_wmma_f32_16x16x4_f32)
#define HAVE_WMMA_F32 1
#endif
#endif

__device__ __forceinline__ v8f wmma4_f32(v2f a, v2f b, v8f c) {
#ifdef HAVE_WMMA_F32
    // (neg_a, A, neg_b, B, c_mod, C, reuse_a, reuse_b)
    return __builtin_amdgcn_wmma_f32_16x16x4_f32(false, a, false, b, (short)0, c,
                                                 false, false);
#else
    (void)a; (void)b;    // host-pass stub
    return c;
#endif
}

// ---------------------------------------------------------------------------
// WMMA GEMM with K fixed at 128:  C[M x Nout] = A[M x 128] * B + bias
// B element (k,n) read as B[k*bRowStride + n*bColStride] (serves row-major
// W1/W2 and transposed W_ih/W_lin). One wave owns one 16-column panel:
// the full 128x16 B panel is preloaded into 32 v2f fragments (64 VGPRs),
// then the wave grid-strides over 16-row tiles of A, issuing 32 WMMAs per
// tile with only one b64 A-load per WMMA. M % 16 == 0, Nout % 64 == 0.
// ---------------------------------------------------------------------------
__global__ void __launch_bounds__(128)
k_gemm16_k128(const float* __restrict__ A, const float* __restrict__ B,
              float* __restrict__ C, int M, int Nout,
              int bRowStride, int bColStride, const float* __restrict__ bias)
{
    const int lane  = threadIdx.x & 31;
    const int wv    = threadIdx.x >> 5;
    const int nBase = (blockIdx.y * 4 + wv) * 16;
    if (nBase >= Nout) return;                 // wave-uniform

    const int l    = lane & 15;
    const int hsel = lane >> 4;                // 0 -> K pair (0,1), 1 -> (2,3)
    const int bn   = nBase + l;

    // Preload B panel fragments (ISA layout: lane l, half hsel holds
    // B[4s + 2*hsel + {0,1}][nBase + l])
    v2f bfrag[32];
#pragma unroll
    for (int s = 0; s < 32; ++s) {
        const int k0 = 4 * s + 2 * hsel;
        bfrag[s].x = B[(size_t)k0 * bRowStride + (size_t)bn * bColStride];
        bfrag[s].y = B[(size_t)(k0 + 1) * bRowStride + (size_t)bn * bColStride];
    }
    const float bv = bias ? bias[bn] : 0.0f;

    for (int mt = blockIdx.x; mt * 16 < M; mt += gridDim.x) {
        const int mBase = mt * 16;
        const float* aRow = A + (size_t)(mBase + l) * 128 + 2 * hsel;
        v8f acc = {};
#pragma unroll
        for (int s = 0; s < 32; ++s) {
            const float2 av = *(const float2*)(aRow + 4 * s);
            v2f a; a.x = av.x; a.y = av.y;
            acc = wmma4_f32(a, bfrag[s], acc);
        }
#pragma unroll
        for (int r = 0; r < 8; ++r) {
            const int row = mBase + r + 8 * hsel;   // C VGPR r -> rows r / r+8
            C[(size_t)row * Nout + bn] = acc[r] + bv;
        }
    }
}

// ---------------------------------------------------------------------------
// GCN normalization: deg (with self-loop) -> dinv = rsqrt(deg)
// ---------------------------------------------------------------------------
__global__ void k_deg_init(float* deg, int N) {
    int i = blockIdx.x * blockDim.x + threadIdx.x;
    if (i < N) deg[i] = 1.0f;                  // self-loop
}
__global__ void k_deg_edges(const long long* __restrict__ dst, float* deg, int E) {
    int e = blockIdx.x * blockDim.x + threadIdx.x;
    if (e < E) atomicAdd(&deg[(int)dst[e]], 1.0f);
}
__global__ void k_dinv(float* deg, int N) {
    int i = blockIdx.x * blockDim.x + threadIdx.x;
    if (i < N) { float d = deg[i]; deg[i] = (d > 0.0f) ? rsqrtf(d) : 0.0f; }
}

// ---------------------------------------------------------------------------
// Aggregation: agg = selfloop + scatter-add over edges (H = 128, float4 lanes)
// ---------------------------------------------------------------------------
__global__ void k_selfloop(const float* __restrict__ xw, const float* __restrict__ dinv,
                           float* __restrict__ agg, int N)
{
    int idx = blockIdx.x * blockDim.x + threadIdx.x;      // N*32 work items
    int i = idx >> 5, q = idx & 31;
    if (i >= N) return;
    float n = dinv[i] * dinv[i];
    float4 v = ((const float4*)(xw + (size_t)i * 128))[q];
    float4 r = make_float4(v.x * n, v.y * n, v.z * n, v.w * n);
    ((float4*)(agg + (size_t)i * 128))[q] = r;
}

__global__ void k_scatter(const float* __restrict__ xw,
                          const long long* __restrict__ src,
                          const long long* __restrict__ dst,
                          const float* __restrict__ dinv,
                          float* __restrict__ agg, int E)
{
    int idx = blockIdx.x * blockDim.x + threadIdx.x;      // E*32 work items
    int e = idx >> 5, q = idx & 31;
    if (e >= E) return;
    int s = (int)src[e], d = (int)dst[e];
    float n = dinv[s] * dinv[d];
    float4 v = ((const float4*)(xw + (size_t)s * 128))[q];
    float* base = agg + (size_t)d * 128 + q * 4;
    atomicAdd(base + 0, v.x * n);
    atomicAdd(base + 1, v.y * n);
    atomicAdd(base + 2, v.z * n);
    atomicAdd(base + 3, v.w * n);
}

__global__ void k_bias_relu(const float* __restrict__ agg, const float* __restrict__ b,
                            float* __restrict__ h, int total)
{
    int idx = blockIdx.x * blockDim.x + threadIdx.x;
    if (idx < total) {
        int f = idx & 127;
        h[idx] = fmaxf(agg[idx] + b[f], 0.0f);
    }
}

// ---------------------------------------------------------------------------
// LSTM helpers
// ---------------------------------------------------------------------------
__global__ void k_add_bias(const float* __restrict__ b_ih, const float* __restrict__ b_hh,
                           float* __restrict__ gbias, int n)
{
    int i = blockIdx.x * blockDim.x + threadIdx.x;
    if (i < n) gbias[i] = b_ih[i] + b_hh[i];
}

__global__ void k_transpose_whh(const float* __restrict__ W, float* __restrict__ WT, int total)
{
    int idx = blockIdx.x * blockDim.x + threadIdx.x;      // 512*128
    if (idx < total) {
        int j = idx >> 7, k = idx & 127;                  // W[j][k], j<512
        WT[k * 512 + j] = W[idx];
    }
}

// Persistent single-workgroup sequential scan. Each of the 512 threads keeps
// its 128-float W_hh row entirely in VGPRs (preloaded coalesced from W_hh^T),
// so the 20000-step inner loop is pure v_fmac against an LDS broadcast of h:
// zero per-step L2 traffic, VALU-bound at ~512 cycles/step on one WGP.
__global__ void __launch_bounds__(512)
k_lstm(const float* __restrict__ gx,   // [N,512] = x@W_ih^T + b_ih + b_hh
       const float* __restrict__ WT,   // [128,512] = W_hh^T
       float* __restrict__ hs, int N)
{
    __shared__ float sh[128];
    __shared__ float sg[512];
    const int j = threadIdx.x;

    float w[128];                      // register-resident recurrent weights
#pragma unroll
    for (int k = 0; k < 128; ++k)
        w[k] = WT[k * 512 + j];        // lane-consecutive -> coalesced

    if (j < 128) sh[j] = 0.0f;
    float c = 0.0f;
    __syncthreads();

    for (int t = 0; t < N; ++t) {
        float acc = gx[(size_t)t * 512 + j];
#pragma unroll
        for (int k = 0; k < 128; ++k)
            acc = fmaf(w[k], sh[k], acc);   // sh[k]: LDS broadcast, no conflicts
        sg[j] = acc;
        __syncthreads();
        if (j < 128) {
            float iv = 1.0f / (1.0f + expf(-sg[j]));
            float fv = 1.0f / (1.0f + expf(-sg[j + 128]));
            float gv = tanhf(sg[j + 256]);
            float ov = 1.0f / (1.0f + expf(-sg[j + 384]));
            c = fv * c + iv * gv;
            float h = ov * tanhf(c);
            sh[j] = h;
            hs[(size_t)t * 128 + j] = h;
        }
        __syncthreads();
    }
}

// ---------------------------------------------------------------------------
// Launch
// ---------------------------------------------------------------------------
extern "C" void kernel_launch(void* const* d_in, const int* in_sizes, int n_in,
                              void* d_out, int out_size, void* d_ws, size_t ws_size,
                              hipStream_t stream)
{
    (void)n_in; (void)out_size; (void)ws_size;
    const int H = 128, G = 512, OUT = 64;
    const int N = in_sizes[0] / H;            // 20000
    const int E = in_sizes[1] / 2;            // 640000

    const float*     x     = (const float*)d_in[0];
    const long long* eidx  = (const long long*)d_in[1];
    const long long* esrc  = eidx;
    const long long* edst  = eidx + E;
    const float*     W1    = (const float*)d_in[2];
    const float*     b1    = (const float*)d_in[3];
    const float*     W2    = (const float*)d_in[4];
    const float*     b2    = (const float*)d_in[5];
    const float*     W_ih  = (const float*)d_in[6];
    const float*     W_hh  = (const float*)d_in[7];
    const float*     b_ih  = (const float*)d_in[8];
    const float*     b_hh  = (const float*)d_in[9];
    const float*     W_lin = (const float*)d_in[10];
    const float*     b_lin = (const float*)d_in[11];
    float*           out   = (float*)d_out;

    // Workspace carve-out (floats)
    float* w = (float*)d_ws;
    size_t off = 0;
    auto carve = [&](size_t nElems) { float* p = w + off; off += (nElems + 255) & ~size_t(255); return p; };
    float* dinv  = carve((size_t)N);
    float* xw    = carve((size_t)N * H);
    float* agg   = carve((size_t)N * H);
    float* h     = carve((size_t)N * H);
    float* gx    = carve((size_t)N * G);
    float* hs    = carve((size_t)N * H);
    float* gbias = carve((size_t)G);
    float* WT    = carve((size_t)G * H);

    const int T256 = 256;
    const int GX   = 256;                      // grid-stride row-tile blocks
    dim3 blkG(128);

    // ---- degree / normalization (shared by both GCN layers) ----
    k_deg_init <<<(N + T256 - 1) / T256, T256, 0, stream>>>(dinv, N);
    k_deg_edges<<<(E + T256 - 1) / T256, T256, 0, stream>>>(edst, dinv, E);
    k_dinv     <<<(N + T256 - 1) / T256, T256, 0, stream>>>(dinv, N);

    // ---- GCN layer 1: xw = x @ W1 ; aggregate ; relu(+b1) -> h ----
    k_gemm16_k128<<<dim3(GX, H / 64), blkG, 0, stream>>>(x, W1, xw, N, H, H, 1, nullptr);
    k_selfloop <<<((size_t)N * 32 + T256 - 1) / T256, T256, 0, stream>>>(xw, dinv, agg, N);
    k_scatter  <<<((size_t)E * 32 + T256 - 1) / T256, T256, 0, stream>>>(xw, esrc, edst, dinv, agg, E);
    k_bias_relu<<<((size_t)N * H + T256 - 1) / T256, T256, 0, stream>>>(agg, b1, h, N * H);

    // ---- GCN layer 2: xw = h @ W2 ; aggregate ; relu(+b2) -> h ----
    k_gemm16_k128<<<dim3(GX, H / 64), blkG, 0, stream>>>(h, W2, xw, N, H, H, 1, nullptr);
    k_selfloop <<<((size_t)N * 32 + T256 - 1) / T256, T256, 0, stream>>>(xw, dinv, agg, N);
    k_scatter  <<<((size_t)E * 32 + T256 - 1) / T256, T256, 0, stream>>>(xw, esrc, edst, dinv, agg, E);
    k_bias_relu<<<((size_t)N * H + T256 - 1) / T256, T256, 0, stream>>>(agg, b2, h, N * H);

    // ---- LSTM: gx = h @ W_ih^T + (b_ih + b_hh), then sequential scan ----
    k_add_bias<<<(G + T256 - 1) / T256, T256, 0, stream>>>(b_ih, b_hh, gbias, G);
    k_gemm16_k128<<<dim3(GX, G / 64), blkG, 0, stream>>>(h, W_ih, gx, N, G, 1, H, gbias);
    k_transpose_whh<<<(G * H + T256 - 1) / T256, T256, 0, stream>>>(W_hh, WT, G * H);
    k_lstm<<<1, 512, 0, stream>>>(gx, WT, hs, N);

    // ---- Final projection: out = hs @ W_lin^T + b_lin ----
    k_gemm16_k128<<<dim3(GX, OUT / 64), blkG, 0, stream>>>(hs, W_lin, out, N, OUT, 1, H, b_lin);
}